// VisionEncoder_88433376625345
// MI455X (gfx1250) — compile-verified
//
#include <hip/hip_runtime.h>
#include <hip/hip_bf16.h>
#include <math.h>

// ---------------- constants from the reference ----------------
#define S_TOT   1152
#define SEGLEN  576
#define NEMB    1024
#define NHEADS  16
#define HDIM    64
#define NMLP    4096
#define HID     4096
#define NOUT    2048
#define MROWS   288          // S_TOT / 4 (MERGE*MERGE)
#define INPATCH 1536
#define GRID    48
#define IMG_HW  24

typedef __attribute__((ext_vector_type(2))) float v2f;
typedef __attribute__((ext_vector_type(8))) float v8f;

__device__ __forceinline__ float gelu_exact(float x) {
    return 0.5f * x * (1.0f + erff(x * 0.70710678118654752f));
}

// =====================================================================
// Generic fp32 WMMA GEMM:  C[M,N] = act(A[M,K] @ W[N,K]^T + bias) (+res)
// One wave -> one 16x16 output tile. 8 waves / block (256 thr).
// grid = (N/128, M/16). All dims multiples of 16; N multiple of 128.
// =====================================================================
__global__ __launch_bounds__(256)
void gemm_f32_wmma(const float* __restrict__ A,
                   const float* __restrict__ W,
                   const float* __restrict__ bias,
                   const float* __restrict__ residual,
                   float* __restrict__ C,
                   int M, int N, int K, int act)
{
    const int wave = threadIdx.x >> 5;
    const int lane = threadIdx.x & 31;
    const int hl   = lane >> 4;       // lane half: selects K pair
    const int lr   = lane & 15;

    const int nBase = (blockIdx.x * 8 + wave) * 16;
    const int mBase = blockIdx.y * 16;
    if (nBase >= N) return;           // wave-uniform

    // A fragment source: row mBase+lr, k offset 2*hl (ISA A layout 16x4 f32)
    const float* __restrict__ arow = A + (size_t)(mBase + lr) * K + 2 * hl;
    // B fragment source: B[k][n] = W[n][k]; col nBase+lr, same k offset
    const float* __restrict__ wrow = W + (size_t)(nBase + lr) * K + 2 * hl;

    v8f acc = {0.f,0.f,0.f,0.f,0.f,0.f,0.f,0.f};
    #pragma unroll 4
    for (int k = 0; k < K; k += 4) {
        v2f a = *(const v2f*)(arow + k);
        v2f b = *(const v2f*)(wrow + k);
        acc = __builtin_amdgcn_wmma_f32_16x16x4_f32(
            false, a, false, b, (short)0, acc, false, false);
    }

    const int col = nBase + lr;
    const float bv = bias ? bias[col] : 0.0f;
    #pragma unroll
    for (int i = 0; i < 8; ++i) {
        const int row = mBase + hl * 8 + i;      // ISA C/D layout
        float v = acc[i] + bv;
        if (act == 1) v = gelu_exact(v);
        if (residual) v += residual[(size_t)row * N + col];
        C[(size_t)row * N + col] = v;
    }
}

// =====================================================================
// LayerNorm over last dim D. One block (256 thr) per row.
// =====================================================================
__global__ __launch_bounds__(256)
void ln_kernel(const float* __restrict__ X, const float* __restrict__ g,
               const float* __restrict__ b, float* __restrict__ Y, int D)
{
    const int row = blockIdx.x;
    const float* __restrict__ x = X + (size_t)row * D;
    float* __restrict__ y = Y + (size_t)row * D;

    float s = 0.f, s2 = 0.f;
    for (int i = threadIdx.x; i < D; i += blockDim.x) {
        float v = x[i]; s += v; s2 += v * v;
    }
    #pragma unroll
    for (int off = 16; off > 0; off >>= 1) {
        s  += __shfl_down(s,  off, 32);
        s2 += __shfl_down(s2, off, 32);
    }
    __shared__ float r0[8], r1[8];
    const int wave = threadIdx.x >> 5, lane = threadIdx.x & 31;
    if (lane == 0) { r0[wave] = s; r1[wave] = s2; }
    __syncthreads();
    if (threadIdx.x == 0) {
        float a = 0.f, c = 0.f;
        for (int i = 0; i < 8; ++i) { a += r0[i]; c += r1[i]; }
        r0[0] = a; r1[0] = c;
    }
    __syncthreads();
    const float mean = r0[0] / (float)D;
    const float var  = r1[0] / (float)D - mean * mean;
    const float rstd = rsqrtf(var + 1e-6f);
    for (int i = threadIdx.x; i < D; i += blockDim.x)
        y[i] = (x[i] - mean) * rstd * g[i] + b[i];
}

// =====================================================================
// Window-perm token decode (shared by pos-embed & RoPE):
//   s in [0,576): j=s&1, i=(s>>1)&1, wb=(s>>2)%12, hb=(s>>2)/12
//   pixel row r = 2*hb+i, col c = 2*wb+j
// =====================================================================
__device__ __forceinline__ void decode_tok(int s, int& r, int& c)
{
    int j = s & 1, i = (s >> 1) & 1;
    int q = s >> 2;
    int wb = q % (IMG_HW / 2), hb = q / (IMG_HW / 2);
    r = 2 * hb + i;
    c = 2 * wb + j;
}

// Bilinear pos-embed interpolation add. grid=(NEMB/256, S_TOT), 256 thr.
__global__ __launch_bounds__(256)
void add_pos_kernel(float* __restrict__ x, const float* __restrict__ table)
{
    const int tok = blockIdx.y;
    const int d   = blockIdx.x * blockDim.x + threadIdx.x;
    int r, c; decode_tok(tok % SEGLEN, r, c);

    const float step = (float)(GRID - 1) / (float)(IMG_HW - 1);  // 47/23
    const float hi = r * step, wi = c * step;
    int hf = (int)floorf(hi), wf = (int)floorf(wi);
    int hc = min(hf + 1, GRID - 1), wc = min(wf + 1, GRID - 1);
    float dh = hi - (float)hf, dw = wi - (float)wf;

    float v = table[((size_t)(hf * GRID + wf)) * NEMB + d] * (1.f - dh) * (1.f - dw)
            + table[((size_t)(hf * GRID + wc)) * NEMB + d] * (1.f - dh) * dw
            + table[((size_t)(hc * GRID + wf)) * NEMB + d] * dh * (1.f - dw)
            + table[((size_t)(hc * GRID + wc)) * NEMB + d] * dh * dw;
    x[(size_t)tok * NEMB + d] += v;
}

// RoPE cos/sin tables (S_TOT x 64). grid=S_TOT, 64 thr.
__global__ __launch_bounds__(64)
void rope_tables(float* __restrict__ cosT, float* __restrict__ sinT)
{
    const int tok = blockIdx.x;
    const int d   = threadIdx.x;            // 0..63
    int r, c; decode_tok(tok % SEGLEN, r, c);
    const int d0 = d & 31;                  // tile(...,(1,2))
    float ang;
    if (d0 < 16) ang = (float)r * powf(10000.0f, -(float)d0 / 16.0f);
    else         ang = (float)c * powf(10000.0f, -(float)(d0 - 16) / 16.0f);
    cosT[(size_t)tok * HDIM + d] = cosf(ang);
    sinT[(size_t)tok * HDIM + d] = sinf(ang);
}

// In-place RoPE on q (mat 0) and k (mat 1) of qkv[S,3,16,64].
// grid=(S_TOT,2), 512 thr (head = tid>>5, d = tid&31 handles pair d,d+32).
__global__ __launch_bounds__(512)
void rope_apply(float* __restrict__ qkv,
                const float* __restrict__ cosT, const float* __restrict__ sinT)
{
    const int tok = blockIdx.x, mat = blockIdx.y;
    const int head = threadIdx.x >> 5, d = threadIdx.x & 31;
    float* base = qkv + ((size_t)tok * 3 + mat) * NEMB + head * HDIM;
    const float c0 = cosT[(size_t)tok * HDIM + d];
    const float s0 = sinT[(size_t)tok * HDIM + d];
    const float c1 = cosT[(size_t)tok * HDIM + d + 32];
    const float s1 = sinT[(size_t)tok * HDIM + d + 32];
    const float x0 = base[d], x1 = base[d + 32];
    base[d]      = x0 * c0 - x1 * s0;   // rotate_half: -x[d+32] for d<32
    base[d + 32] = x1 * c1 + x0 * s1;   //  x[d-32]    for d>=32
}

// =====================================================================
// Attention: one wave per (16-query tile, head, segment).
// scores (WMMA) -> LDS -> softmax (in-wave) -> P·V (WMMA) -> out[S,NEMB]
// =====================================================================
__global__ __launch_bounds__(32)
void attn_kernel(const float* __restrict__ qkv, float* __restrict__ out)
{
    __shared__ float sc[16][SEGLEN + 1];   // +1 pad: kill bank conflicts

    const int lane = threadIdx.x & 31;
    const int hl = lane >> 4, lr = lane & 15;
    const int qt = blockIdx.x, head = blockIdx.y, seg = blockIdx.z;
    const int q0 = seg * SEGLEN + qt * 16;
    const size_t stT = 3 * NEMB;           // token stride in qkv

    const float* __restrict__ Qb = qkv + 0 * NEMB + head * HDIM;
    const float* __restrict__ Kb = qkv + 1 * NEMB + head * HDIM;
    const float* __restrict__ Vb = qkv + 2 * NEMB + head * HDIM;

    // ---- scores = Q @ K^T / 8 ----
    const float* qrow = Qb + (size_t)(q0 + lr) * stT + 2 * hl;
    for (int nt = 0; nt < SEGLEN / 16; ++nt) {
        const int kb = seg * SEGLEN + nt * 16;
        const float* krow = Kb + (size_t)(kb + lr) * stT + 2 * hl;
        v8f acc = {0.f,0.f,0.f,0.f,0.f,0.f,0.f,0.f};
        #pragma unroll
        for (int k = 0; k < HDIM; k += 4) {
            v2f a = { qrow[k], qrow[k + 1] };
            v2f b = { krow[k], krow[k + 1] };
            acc = __builtin_amdgcn_wmma_f32_16x16x4_f32(
                false, a, false, b, (short)0, acc, false, false);
        }
        #pragma unroll
        for (int i = 0; i < 8; ++i)
            sc[hl * 8 + i][nt * 16 + lr] = acc[i] * 0.125f;  // 1/sqrt(64)
    }
    __syncthreads();

    // ---- softmax over 576 keys; lane lr owns row lr, half hl of cols ----
    {
        const int c0 = hl * (SEGLEN / 2);
        float mx = -3.402823466e38f;
        for (int j = 0; j < SEGLEN / 2; ++j) mx = fmaxf(mx, sc[lr][c0 + j]);
        mx = fmaxf(mx, __shfl_xor(mx, 16, 32));
        float sum = 0.f;
        for (int j = 0; j < SEGLEN / 2; ++j) {
            float e = expf(sc[lr][c0 + j] - mx);
            sc[lr][c0 + j] = e; sum += e;
        }
        sum += __shfl_xor(sum, 16, 32);
        const float inv = 1.0f / sum;
        for (int j = 0; j < SEGLEN / 2; ++j) sc[lr][c0 + j] *= inv;
    }
    __syncthreads();

    // ---- out = P @ V ----
    for (int dt = 0; dt < HDIM / 16; ++dt) {
        v8f acc = {0.f,0.f,0.f,0.f,0.f,0.f,0.f,0.f};
        for (int k = 0; k < SEGLEN; k += 4) {
            v2f a = { sc[lr][k + 2 * hl], sc[lr][k + 2 * hl + 1] };
            const float* vp = Vb + (size_t)(seg * SEGLEN + k + 2 * hl) * stT
                            + dt * 16 + lr;
            v2f b = { vp[0], vp[stT] };
            acc = __builtin_amdgcn_wmma_f32_16x16x4_f32(
                false, a, false, b, (short)0, acc, false, false);
        }
        #pragma unroll
        for (int i = 0; i < 8; ++i)
            out[(size_t)(q0 + hl * 8 + i) * NEMB + head * HDIM + dt * 16 + lr]
                = acc[i];
    }
}

// =====================================================================
// Host side
// =====================================================================
static inline void gemm(const float* A, const float* W, const float* bias,
                        const float* res, float* C, int M, int N, int K,
                        int act, hipStream_t s)
{
    dim3 g(N / 128, M / 16);
    gemm_f32_wmma<<<g, 256, 0, s>>>(A, W, bias, res, C, M, N, K, act);
}

struct Blk { const float *ln1_g,*ln1_b,*ln2_g,*ln2_b,*qkv_w,*qkv_b,
                         *proj_w,*proj_b,*fc1_w,*fc1_b,*fc2_w,*fc2_b; };
struct Mrg { const float *g,*b,*fc1_w,*fc1_b,*fc2_w,*fc2_b; };

extern "C" void kernel_launch(void* const* d_in, const int* in_sizes, int n_in,
                              void* d_out, int out_size, void* d_ws, size_t ws_size,
                              hipStream_t stream)
{
    (void)in_sizes; (void)n_in; (void)out_size; (void)ws_size;
    const float* pixels    = (const float*)d_in[0];
    // d_in[1] = d_image (unused; geometry is compile-time constant)
    const float* patch_w   = (const float*)d_in[2];
    const float* patch_b   = (const float*)d_in[3];
    const float* pos_embed = (const float*)d_in[4];

    int p = 5;
    Blk blk[2];
    for (int i = 0; i < 2; ++i) {
        blk[i].ln1_g  = (const float*)d_in[p++]; blk[i].ln1_b  = (const float*)d_in[p++];
        blk[i].ln2_g  = (const float*)d_in[p++]; blk[i].ln2_b  = (const float*)d_in[p++];
        blk[i].qkv_w  = (const float*)d_in[p++]; blk[i].qkv_b  = (const float*)d_in[p++];
        blk[i].proj_w = (const float*)d_in[p++]; blk[i].proj_b = (const float*)d_in[p++];
        blk[i].fc1_w  = (const float*)d_in[p++]; blk[i].fc1_b  = (const float*)d_in[p++];
        blk[i].fc2_w  = (const float*)d_in[p++]; blk[i].fc2_b  = (const float*)d_in[p++];
    }
    Mrg mrg[3]; // [0]=final merger, [1]=deepstack0, [2]=deepstack1
    for (int i = 0; i < 3; ++i) {
        mrg[i].g     = (const float*)d_in[p++]; mrg[i].b     = (const float*)d_in[p++];
        mrg[i].fc1_w = (const float*)d_in[p++]; mrg[i].fc1_b = (const float*)d_in[p++];
        mrg[i].fc2_w = (const float*)d_in[p++]; mrg[i].fc2_b = (const float*)d_in[p++];
    }

    float* ws = (float*)d_ws;
    float* x      = ws;                                  // 1152*1024
    float* lnbuf  = x      + (size_t)S_TOT * NEMB;       // 1152*1024
    float* qkv    = lnbuf  + (size_t)S_TOT * NEMB;       // 1152*3072
    float* aout   = qkv    + (size_t)S_TOT * 3 * NEMB;   // 1152*1024
    float* h      = aout   + (size_t)S_TOT * NEMB;       // 1152*4096
    float* mbuf   = h      + (size_t)S_TOT * NMLP;       // 288*4096
    float* mbuf2  = mbuf   + (size_t)MROWS * HID;        // 288*4096
    float* cosT   = mbuf2  + (size_t)MROWS * HID;        // 1152*64
    float* sinT   = cosT   + (size_t)S_TOT * HDIM;       // 1152*64

    float* outp = (float*)d_out;
    const size_t OUTSZ = (size_t)MROWS * NOUT;           // 589824

    // 1) patch embed + pos embed
    gemm(pixels, patch_w, patch_b, nullptr, x, S_TOT, NEMB, INPATCH, 0, stream);
    add_pos_kernel<<<dim3(NEMB / 256, S_TOT), 256, 0, stream>>>(x, pos_embed);
    rope_tables<<<S_TOT, 64, 0, stream>>>(cosT, sinT);

    // 2) transformer blocks + deepstack mergers
    for (int L = 0; L < 2; ++L) {
        const Blk& B = blk[L];
        ln_kernel<<<S_TOT, 256, 0, stream>>>(x, B.ln1_g, B.ln1_b, lnbuf, NEMB);
        gemm(lnbuf, B.qkv_w, B.qkv_b, nullptr, qkv, S_TOT, 3 * NEMB, NEMB, 0, stream);
        rope_apply<<<dim3(S_TOT, 2), 512, 0, stream>>>(qkv, cosT, sinT);
        attn_kernel<<<dim3(SEGLEN / 16, NHEADS, 2), 32, 0, stream>>>(qkv, aout);
        gemm(aout, B.proj_w, B.proj_b, x, x, S_TOT, NEMB, NEMB, 0, stream);

        ln_kernel<<<S_TOT, 256, 0, stream>>>(x, B.ln2_g, B.ln2_b, lnbuf, NEMB);
        gemm(lnbuf, B.fc1_w, B.fc1_b, nullptr, h, S_TOT, NMLP, NEMB, 1, stream);
        gemm(h, B.fc2_w, B.fc2_b, x, x, S_TOT, NEMB, NMLP, 0, stream);

        // deepstack merger (postshuffle=True): LN over HID on reshaped x
        const Mrg& D = mrg[1 + L];
        ln_kernel<<<MROWS, 256, 0, stream>>>(x, D.g, D.b, mbuf, HID);
        gemm(mbuf, D.fc1_w, D.fc1_b, nullptr, mbuf2, MROWS, HID, HID, 1, stream);
        gemm(mbuf2, D.fc2_w, D.fc2_b, nullptr, outp + (1 + L) * OUTSZ,
             MROWS, NOUT, HID, 0, stream);
    }

    // 3) final merger (postshuffle=False): LN over NEMB, then reshape
    ln_kernel<<<S_TOT, 256, 0, stream>>>(x, mrg[0].g, mrg[0].b, lnbuf, NEMB);
    gemm(lnbuf, mrg[0].fc1_w, mrg[0].fc1_b, nullptr, mbuf2, MROWS, HID, HID, 1, stream);
    gemm(mbuf2, mrg[0].fc2_w, mrg[0].fc2_b, nullptr, outp, MROWS, NOUT, HID, 0, stream);
}